// NeuralMemory_31026843746848
// MI455X (gfx1250) — compile-verified
//
#include <hip/hip_runtime.h>

typedef __attribute__((ext_vector_type(2))) float v2f;
typedef __attribute__((ext_vector_type(8))) float v8f;

#define TOK   4096     // B*N
#define BN    1024     // N (tokens per batch)
#define DD    128      // D

__device__ __forceinline__ float sigf(float x)  { return 1.0f / (1.0f + __expf(-x)); }
__device__ __forceinline__ float siluf(float x) { return x * sigf(x); }
__device__ __forceinline__ float dsiluf(float x){ float s = sigf(x); return s * (1.0f + x * (1.0f - s)); }

// ---------------------------------------------------------------------------
// Generic f32 WMMA GEMM:  C[m, n0:n0+128] = op(A) @ (W or W^T)  (K = 128)
//   - 256 threads = 8 waves; block computes 32 rows x 128 cols
//   - wave (wm=w>>2, wn=w&3) computes rows [m0,m0+16) x cols [32*wn, +32) (2 tiles)
//   - wTrans: use W^T (for backward deltas)
//   - Zmul:   multiply result elementwise by silu'(Zmul) (ld fixed = 128)
//   - Cout:   raw result (nullable); SiluOut: silu(result) (nullable)
//   - wBatchStride: per-batch weights (batch = row/1024), 0 = shared
// WMMA fragment layouts per CDNA5 ISA 7.12.2 (32-bit A 16x4, 32-bit C/D 16x16).
// ---------------------------------------------------------------------------
__global__ __launch_bounds__(256) void gemm128(
    const float* __restrict__ A, int lda,
    const float* __restrict__ W, int ldw, int wBatchStride, int wTrans,
    const float* __restrict__ Zmul,
    float* __restrict__ Cout, int ldc,
    float* __restrict__ SiluOut, int ldso)
{
  const int wave = threadIdx.x >> 5;
  const int lane = threadIdx.x & 31;
  const int half = lane >> 4;        // 0: K/M low half, 1: high half per ISA layout
  const int l    = lane & 15;
  const int m0   = blockIdx.x * 32 + (wave >> 2) * 16;
  const int n0   = blockIdx.y * 128 + (wave & 3) * 32;
  const int batch = (blockIdx.x * 32) / BN;
  const float* Wb = W + (size_t)batch * wBatchStride;

  v8f acc0 = {};
  v8f acc1 = {};
  const float* arow = A + (size_t)(m0 + l) * lda + 2 * half;

  if (wTrans) {
    // B frag[k][n] = W[n][k] : contiguous pair per lane
    const float* wr0 = Wb + (size_t)(n0 + l) * ldw + 2 * half;
    const float* wr1 = Wb + (size_t)(n0 + 16 + l) * ldw + 2 * half;
    for (int k0 = 0; k0 < DD; k0 += 4) {
      v2f a  = *(const v2f*)(arow + k0);
      v2f b0 = *(const v2f*)(wr0 + k0);
      v2f b1 = *(const v2f*)(wr1 + k0);
      acc0 = __builtin_amdgcn_wmma_f32_16x16x4_f32(false, a, false, b0, (short)0, acc0, false, false);
      acc1 = __builtin_amdgcn_wmma_f32_16x16x4_f32(false, a, false, b1, (short)0, acc1, false, false);
    }
  } else {
    for (int k0 = 0; k0 < DD; k0 += 4) {
      v2f a = *(const v2f*)(arow + k0);
      const float* r0 = Wb + (size_t)(k0 + 2 * half) * ldw + n0 + l; // row k0+2h
      const float* r1 = r0 + ldw;                                    // row k0+2h+1
      v2f b0, b1;
      b0.x = r0[0];  b0.y = r1[0];
      b1.x = r0[16]; b1.y = r1[16];
      acc0 = __builtin_amdgcn_wmma_f32_16x16x4_f32(false, a, false, b0, (short)0, acc0, false, false);
      acc1 = __builtin_amdgcn_wmma_f32_16x16x4_f32(false, a, false, b1, (short)0, acc1, false, false);
    }
  }

  // Epilogue: lane l, vgpr r holds C[m0 + r + 8*half][n0 (+16) + l]
  #pragma unroll
  for (int r = 0; r < 8; ++r) {
    const int m = m0 + r + 8 * half;
    float c0 = acc0[r];
    float c1 = acc1[r];
    if (Zmul) {
      c0 *= dsiluf(Zmul[(size_t)m * DD + (n0 + l)]);
      c1 *= dsiluf(Zmul[(size_t)m * DD + (n0 + 16 + l)]);
    }
    if (Cout) {
      Cout[(size_t)m * ldc + n0 + l]      = c0;
      Cout[(size_t)m * ldc + n0 + 16 + l] = c1;
    }
    if (SiluOut) {
      SiluOut[(size_t)m * ldso + n0 + l]      = siluf(c0);
      SiluOut[(size_t)m * ldso + n0 + 16 + l] = siluf(c1);
    }
  }
}

// ---------------------------------------------------------------------------
// Gradient GEMM: U[b] = Wbase + Act[b]^T @ Dlt[b]   (128 x 1024 @ 1024 x 128)
// grid = (4, 1, 4): x tiles the 128 output rows (d), z = batch.
// ---------------------------------------------------------------------------
__global__ __launch_bounds__(256) void gradU(
    const float* __restrict__ Act, int ldact,
    const float* __restrict__ Dlt,
    const float* __restrict__ Wbase,
    float* __restrict__ Uout)
{
  const int wave = threadIdx.x >> 5;
  const int lane = threadIdx.x & 31;
  const int half = lane >> 4;
  const int l    = lane & 15;
  const int m0   = blockIdx.x * 32 + (wave >> 2) * 16;   // output row d
  const int n0   = (wave & 3) * 32;                      // output col e
  const int batch = blockIdx.z;
  const float* Ab = Act + (size_t)batch * BN * ldact + m0 + l;  // column d=m0+l
  const float* Db = Dlt + (size_t)batch * BN * DD;

  v8f acc0 = {};
  v8f acc1 = {};
  for (int k0 = 0; k0 < BN; k0 += 4) {
    const int kr = k0 + 2 * half;
    v2f a;                                   // A frag[m=d][k=n] = Act[n][d]
    a.x = Ab[(size_t)(kr + 0) * ldact];
    a.y = Ab[(size_t)(kr + 1) * ldact];
    const float* d0 = Db + (size_t)kr * DD + n0 + l;
    const float* d1 = d0 + DD;
    v2f b0, b1;                              // B frag[k=n][e] = Dlt[n][e]
    b0.x = d0[0];  b0.y = d1[0];
    b1.x = d0[16]; b1.y = d1[16];
    acc0 = __builtin_amdgcn_wmma_f32_16x16x4_f32(false, a, false, b0, (short)0, acc0, false, false);
    acc1 = __builtin_amdgcn_wmma_f32_16x16x4_f32(false, a, false, b1, (short)0, acc1, false, false);
  }

  #pragma unroll
  for (int r = 0; r < 8; ++r) {
    const int d = m0 + r + 8 * half;
    float* urow = Uout + (size_t)batch * DD * DD + (size_t)d * DD;
    const float* wrow = Wbase + (size_t)d * DD;
    urow[n0 + l]      = wrow[n0 + l]      + acc0[r];
    urow[n0 + 16 + l] = wrow[n0 + 16 + l] + acc1[r];
  }
}

// D3 = (2/D) * (P - V), V lives in KV[:, 128:256]
__global__ __launch_bounds__(256) void d3_elem(
    const float* __restrict__ P, const float* __restrict__ KV, float* __restrict__ D3)
{
  const int idx = blockIdx.x * 256 + threadIdx.x;   // [0, 4096*128)
  const int m = idx >> 7;
  const int n = idx & 127;
  const float v = KV[(size_t)m * 256 + DD + n];
  D3[idx] = (2.0f / 128.0f) * (P[idx] - v);
}

extern "C" void kernel_launch(void* const* d_in, const int* in_sizes, int n_in,
                              void* d_out, int out_size, void* d_ws, size_t ws_size,
                              hipStream_t stream)
{
  const float* S   = (const float*)d_in[0];
  const float* w0  = (const float*)d_in[1];
  const float* w1  = (const float*)d_in[2];
  const float* w2  = (const float*)d_in[3];
  const float* w3  = (const float*)d_in[4];
  const float* wq  = (const float*)d_in[5];
  const float* wkv = (const float*)d_in[6];
  float* out = (float*)d_out;

  // Scratch layout (floats), ~27 MB total
  float* p = (float*)d_ws;
  const size_t T = (size_t)TOK * DD;
  float* KV = p; p += (size_t)TOK * 256;   // K = cols 0..127, V = cols 128..255
  float* Z0 = p; p += T;  float* A1 = p; p += T;
  float* Z1 = p; p += T;  float* A2 = p; p += T;
  float* Z2 = p; p += T;  float* A3 = p; p += T;
  float* P_ = p; p += T;
  float* Da = p; p += T;                   // holds D3, then D1
  float* Db = p; p += T;                   // holds D2, then D0
  float* Xa = p; p += T;
  float* Xb = p; p += T;
  float* U0 = p; p += 4 * DD * DD;
  float* U1 = p; p += 4 * DD * DD;
  float* U2 = p; p += 4 * DD * DD;
  float* U3 = p; p += 4 * DD * DD;

  const dim3 blk(256);
  const dim3 gKV(TOK / 32, 2);   // N = 256
  const dim3 g(TOK / 32, 1);     // N = 128
  const dim3 gg(4, 1, 4);        // grad: 128 rows tiled by 32, 4 batches

  // ---- key/value projection ----
  gemm128<<<gKV, blk, 0, stream>>>(S, DD, wkv, 256, 0, 0, nullptr, KV, 256, nullptr, 0);

  // ---- forward MLP on keys (save Z and silu(Z)) ----
  gemm128<<<g, blk, 0, stream>>>(KV, 256, w0, DD, 0, 0, nullptr, Z0, DD, A1, DD);
  gemm128<<<g, blk, 0, stream>>>(A1, DD, w1, DD, 0, 0, nullptr, Z1, DD, A2, DD);
  gemm128<<<g, blk, 0, stream>>>(A2, DD, w2, DD, 0, 0, nullptr, Z2, DD, A3, DD);
  gemm128<<<g, blk, 0, stream>>>(A3, DD, w3, DD, 0, 0, nullptr, P_, DD, nullptr, 0);

  // ---- output-layer delta ----
  d3_elem<<<dim3((TOK * DD) / 256), blk, 0, stream>>>(P_, KV, Da);

  // ---- per-batch weight updates + backprop deltas ----
  gradU<<<gg, blk, 0, stream>>>(A3, DD, Da, w3, U3);                               // U3 = w3 + A3^T D3
  gemm128<<<g, blk, 0, stream>>>(Da, DD, w3, DD, 0, 1, Z2, Db, DD, nullptr, 0);    // D2 = (D3 w3^T)*silu'(Z2)
  gradU<<<gg, blk, 0, stream>>>(A2, DD, Db, w2, U2);
  gemm128<<<g, blk, 0, stream>>>(Db, DD, w2, DD, 0, 1, Z1, Da, DD, nullptr, 0);    // D1
  gradU<<<gg, blk, 0, stream>>>(A1, DD, Da, w1, U1);
  gemm128<<<g, blk, 0, stream>>>(Da, DD, w1, DD, 0, 1, Z0, Db, DD, nullptr, 0);    // D0
  gradU<<<gg, blk, 0, stream>>>(KV, 256, Db, w0, U0);                              // U0 = w0 + K^T D0

  // ---- query path through updated per-batch weights ----
  gemm128<<<g, blk, 0, stream>>>(S,  DD, wq, DD, 0,       0, nullptr, Xa, DD, nullptr, 0);  // X0
  gemm128<<<g, blk, 0, stream>>>(Xa, DD, U0, DD, DD * DD, 0, nullptr, nullptr, 0, Xb, DD);  // silu(X0 U0[b])
  gemm128<<<g, blk, 0, stream>>>(Xb, DD, U1, DD, DD * DD, 0, nullptr, nullptr, 0, Xa, DD);  // silu(.. U1[b])
  gemm128<<<g, blk, 0, stream>>>(Xa, DD, U2, DD, DD * DD, 0, nullptr, nullptr, 0, Xb, DD);  // silu(.. U2[b])
  gemm128<<<g, blk, 0, stream>>>(Xb, DD, U3, DD, DD * DD, 0, nullptr, out, DD, nullptr, 0); // final (no act)
}